// RelationAwareness_13511967113421
// MI455X (gfx1250) — compile-verified
//
#include <hip/hip_runtime.h>

typedef __attribute__((ext_vector_type(2))) float v2f;
typedef __attribute__((ext_vector_type(8))) float v8f;

#define HEADS 6
#define EXP   10
#define NN1   62
#define NN2   10
#define NTOT  72     // N1 + N2
#define NPAD  80     // 5 tiles of 16

// One wave (32 lanes) per (b, h) slice.
// Phase 1: Wh1[n], Wh2[n] for n in [0,72) into LDS (padded to 80 with zeros).
// Phase 2: e[i,j] = Wh1[i] + Wh2[j] via V_WMMA_F32_16X16X4_F32 rank-2 GEMM:
//          A = [Wh1 | 1 | 0 | 0] (16x4), B = [1 ; Wh2 ; 0 ; 0] (4x16)
//          -> D[i,j] = Wh1[i]*1 + 1*Wh2[j]  (exact fp32 add)
__global__ __launch_bounds__(32) void relation_awareness_kernel(
    const float* __restrict__ feature,   // [B, 62, 5]
    const float* __restrict__ location,  // [B, 62, 3]
    const float* __restrict__ eye,       // [B, 10, 10]
    const float* __restrict__ W_data,    // [60, 5]
    const float* __restrict__ b_data,    // [60]
    const float* __restrict__ W_loc,     // [60, 3]
    const float* __restrict__ b_loc,     // [60]
    const float* __restrict__ W_eye,     // [60, 10]
    const float* __restrict__ b_eye,     // [60]
    const float* __restrict__ a,         // [20]
    float* __restrict__ out)             // [B, 6, 72, 72]
{
    __shared__ float sWh1[NPAD];
    __shared__ float sWh2[NPAD];

    const int lane = threadIdx.x;       // 0..31
    const int bh   = blockIdx.x;        // b*HEADS + h
    const int b    = bh / HEADS;
    const int h    = bh % HEADS;

    // Attention vector halves (wave-uniform -> scalar loads).
    float a1[EXP], a2[EXP];
    #pragma unroll
    for (int d = 0; d < EXP; ++d) { a1[d] = a[d]; a2[d] = a[EXP + d]; }

    // ---- Phase 1: per-row embed + relu + projection onto a ----
    for (int n = lane; n < NPAD; n += 32) {
        float wh1 = 0.f, wh2 = 0.f;
        if (n < NN1) {
            const float* fp = feature  + ((size_t)b * NN1 + n) * 5;
            const float* lp = location + ((size_t)b * NN1 + n) * 3;
            float f0 = fp[0], f1 = fp[1], f2 = fp[2], f3 = fp[3], f4 = fp[4];
            float l0 = lp[0], l1 = lp[1], l2 = lp[2];
            #pragma unroll
            for (int d = 0; d < EXP; ++d) {
                const int c = h * EXP + d;
                float s = b_data[c] + b_loc[c];
                s += f0 * W_data[c * 5 + 0];
                s += f1 * W_data[c * 5 + 1];
                s += f2 * W_data[c * 5 + 2];
                s += f3 * W_data[c * 5 + 3];
                s += f4 * W_data[c * 5 + 4];
                s += l0 * W_loc[c * 3 + 0];
                s += l1 * W_loc[c * 3 + 1];
                s += l2 * W_loc[c * 3 + 2];
                s = fmaxf(s, 0.f);
                wh1 += s * a1[d];
                wh2 += s * a2[d];
            }
        } else if (n < NTOT) {
            const int m = n - NN1;
            const float* ep = eye + ((size_t)b * NN2 + m) * NN2;
            float ev[NN2];
            #pragma unroll
            for (int k = 0; k < NN2; ++k) ev[k] = ep[k];
            #pragma unroll
            for (int d = 0; d < EXP; ++d) {
                const int c = h * EXP + d;
                float s = b_eye[c];
                #pragma unroll
                for (int k = 0; k < NN2; ++k) s += ev[k] * W_eye[c * NN2 + k];
                s = fmaxf(s, 0.f);
                wh1 += s * a1[d];
                wh2 += s * a2[d];
            }
        }
        sWh1[n] = wh1;   // rows 72..79 padded with 0
        sWh2[n] = wh2;
    }
    __syncthreads();

    // ---- Phase 2: 5x5 grid of 16x16 WMMA tiles ----
    const int  m16  = lane & 15;
    const bool lo   = lane < 16;                 // lanes 16..31 hold K=2,3 -> zero
    const float ones = lo ? 1.0f : 0.0f;

    // Hoist ALL fragment LDS loads unconditionally (no predication),
    // then zero the hi half with plain selects (v_cndmask).
    float t1[5], t2[5];
    #pragma unroll
    for (int t = 0; t < 5; ++t) {
        t1[t] = sWh1[t * 16 + m16];
        t2[t] = sWh2[t * 16 + m16];
    }
    float wh1t[5], wh2t[5];
    #pragma unroll
    for (int t = 0; t < 5; ++t) {
        wh1t[t] = lo ? t1[t] : 0.0f;
        wh2t[t] = lo ? t2[t] : 0.0f;
    }

    float* __restrict__ slice = out + (size_t)bh * (NTOT * NTOT);
    // Per-lane store offset: hi half writes rows +8; column = m16.
    const int laneoff = (lo ? 0 : 8) * NTOT + m16;

    #pragma unroll
    for (int ti = 0; ti < 5; ++ti) {
        v2f A;
        A[0] = wh1t[ti];                         // K=0 : Wh1[row] (hi half = 0)
        A[1] = ones;                             // K=1 : ones column
        #pragma unroll
        for (int tj = 0; tj < 5; ++tj) {
            v2f Bm;
            Bm[0] = ones;                        // K=0 row : ones
            Bm[1] = wh2t[tj];                    // K=1 row : Wh2[col]
            v8f cz = {};
            v8f d = __builtin_amdgcn_wmma_f32_16x16x4_f32(
                false, A, false, Bm, (short)0, cz, false, false);

            // Uniform tile base -> SGPR; lane part -> one VGPR; v*72 -> inst offset.
            float* __restrict__ tb = slice + ti * (16 * NTOT) + tj * 16;
            // Static edge analysis: only ti==4 needs a row mask (lanes<16 valid),
            // only tj==4 needs a col mask (m16<8 valid). Interior: folds to true.
            const bool ok = (ti < 4 || lo) && (tj < 4 || m16 < 8);
            if (ok) {
                #pragma unroll
                for (int v = 0; v < 8; ++v)
                    tb[laneoff + v * NTOT] = d[v];
            }
        }
    }
}

extern "C" void kernel_launch(void* const* d_in, const int* in_sizes, int n_in,
                              void* d_out, int out_size, void* d_ws, size_t ws_size,
                              hipStream_t stream) {
    const float* feature  = (const float*)d_in[0];
    const float* location = (const float*)d_in[1];
    const float* eye      = (const float*)d_in[2];
    const float* W_data   = (const float*)d_in[3];
    const float* b_data   = (const float*)d_in[4];
    const float* W_loc    = (const float*)d_in[5];
    const float* b_loc    = (const float*)d_in[6];
    const float* W_eye    = (const float*)d_in[7];
    const float* b_eye    = (const float*)d_in[8];
    const float* a        = (const float*)d_in[9];
    float* out = (float*)d_out;

    const int B = in_sizes[0] / (NN1 * 5);           // 1024
    dim3 grid(B * HEADS);                            // one wave per (b,h) slice
    dim3 block(32);
    relation_awareness_kernel<<<grid, block, 0, stream>>>(
        feature, location, eye, W_data, b_data, W_loc, b_loc, W_eye, b_eye, a, out);
}